// RelativeDifferenceLoss_35613868818814
// MI455X (gfx1250) — compile-verified
//
#include <hip/hip_runtime.h>
#include <hip/hip_bf16.h>

typedef float v2f __attribute__((ext_vector_type(2)));
typedef float v8f __attribute__((ext_vector_type(8)));

#define RDL_WAVES 16
#define RDL_THREADS (RDL_WAVES * 32)

// Single-workgroup closed-form reduction:
//   loss = 2*(N*sum(e^2) - sum(e)^2) / (N*(N-1)),  e = pred - tgt
// Accumulation runs through V_WMMA_F32_16X16X4_F32 with an all-ones B matrix:
// each WMMA folds 64 f32 values (2 A-VGPRs across 32 lanes) into the 16x16
// f32 accumulator. Two chains: one for e, one for e^2.
__global__ __launch_bounds__(RDL_THREADS) void
RelativeDifferenceLoss_35613868818814_kernel(const float* __restrict__ pred,
                                             const float* __restrict__ tgt,
                                             float* __restrict__ out, int n) {
    const int tid  = threadIdx.x;
    const int wave = tid >> 5;
    const int lane = tid & 31;

    v8f acc1 = {};  // sum(e)
    v8f acc2 = {};  // sum(e^2)
    v2f ones;
    ones[0] = 1.0f;
    ones[1] = 1.0f;

    const int per_wave = n / RDL_WAVES;   // 1024 for N=16384
    const int iters    = per_wave / 128;  // each iter: 32 lanes * 4 floats = 128 elems
    const int base     = wave * per_wave + lane * 4;

    for (int i = 0; i < iters; ++i) {
        const int idx = base + i * 128;
        const float4 p = *reinterpret_cast<const float4*>(pred + idx);
        const float4 t = *reinterpret_cast<const float4*>(tgt + idx);
        const float e0 = p.x - t.x;
        const float e1 = p.y - t.y;
        const float e2 = p.z - t.z;
        const float e3 = p.w - t.w;

        v2f a0; a0[0] = e0;      a0[1] = e1;
        v2f a1; a1[0] = e2;      a1[1] = e3;
        v2f q0; q0[0] = e0 * e0; q0[1] = e1 * e1;
        v2f q1; q1[0] = e2 * e2; q1[1] = e3 * e3;

        // D[m][n] = sum_k A[m][k] + C[m][n]  (B == ones)
        acc1 = __builtin_amdgcn_wmma_f32_16x16x4_f32(false, a0, false, ones,
                                                     (short)0, acc1, false, false);
        acc1 = __builtin_amdgcn_wmma_f32_16x16x4_f32(false, a1, false, ones,
                                                     (short)0, acc1, false, false);
        acc2 = __builtin_amdgcn_wmma_f32_16x16x4_f32(false, q0, false, ones,
                                                     (short)0, acc2, false, false);
        acc2 = __builtin_amdgcn_wmma_f32_16x16x4_f32(false, q1, false, ones,
                                                     (short)0, acc2, false, false);
    }

    // Every column n of D is identical. Per-lane sum over the 8 C/D VGPRs:
    // lanes 0..15 hold sum over m=0..7, lanes 16..31 hold sum over m=8..15.
    float s1 = acc1[0] + acc1[1] + acc1[2] + acc1[3] +
               acc1[4] + acc1[5] + acc1[6] + acc1[7];
    float s2 = acc2[0] + acc2[1] + acc2[2] + acc2[3] +
               acc2[4] + acc2[5] + acc2[6] + acc2[7];

    const float w1 = __shfl(s1, 0, 32) + __shfl(s1, 16, 32);
    const float w2 = __shfl(s2, 0, 32) + __shfl(s2, 16, 32);

    __shared__ float sh1[RDL_WAVES];
    __shared__ float sh2[RDL_WAVES];
    if (lane == 0) {
        sh1[wave] = w1;
        sh2[wave] = w2;
    }
    __syncthreads();

    if (tid == 0) {
        float S1 = 0.0f, S2 = 0.0f;
        for (int w = 0; w < RDL_WAVES; ++w) {
            S1 += sh1[w];
            S2 += sh2[w];
        }
        const float nf = (float)n;
        out[0] = 2.0f * (nf * S2 - S1 * S1) / (nf * (nf - 1.0f));
    }
}

extern "C" void kernel_launch(void* const* d_in, const int* in_sizes, int n_in,
                              void* d_out, int out_size, void* d_ws, size_t ws_size,
                              hipStream_t stream) {
    const float* pred = (const float*)d_in[0];
    const float* tgt  = (const float*)d_in[1];
    float* out        = (float*)d_out;
    const int n       = in_sizes[0];  // 16384

    RelativeDifferenceLoss_35613868818814_kernel<<<1, RDL_THREADS, 0, stream>>>(
        pred, tgt, out, n);
}